// Cutout_47588237639902
// MI455X (gfx1250) — compile-verified
//
#include <hip/hip_runtime.h>
#include <hip/hip_bf16.h>

// Cutout: out[b,c,h,w] = x[b,c,h,w], zeroed inside a 16x16 box centered at
// (cy[b], cx[b]). Pure streaming: 154 MB read + 154 MB write ~= 13 us at
// 23.3 TB/s HBM. 0 FLOP/byte -> WMMA not applicable; optimize the memory path.
//
// gfx1250 strategy:
//  - block(56,4): threadIdx.x == w4 -> no %56; h = r%224 via two conditional
//    subtracts. 224 threads = 7 wave32 waves; dense b128 coalescing.
//  - Uniform per-image base pointer (SGPR) + 32-bit unsigned lane offset
//    -> GVS addressing (global_load_b128 saddr + voffset), no 64-bit VALU adds.
//  - blockIdx.y = b: cy[b]/cx[b] block-uniform -> s_load_b32 + SALU.
//  - x loaded regular-temporal (fits 192 MB L2 across graph replays); output
//    stored non-temporal (th:TH_NT) so the write stream doesn't evict x.
//  - anyCov guard: only waves actually intersecting the 16x16 box execute the
//    4 per-component cndmask selects; everyone else is a straight b128 copy.

typedef float v4f __attribute__((ext_vector_type(4)));

#define LENGTH_HALF 8        // LENGTH = 16
#define W4_PER_ROW  56       // 224 / 4
#define ROWS_PER_IMG 672     // C*H = 3*224
#define ROWS_PER_BLK 4
#define F4_PER_IMG  (ROWS_PER_IMG * W4_PER_ROW)  // 37632

__global__ __launch_bounds__(224) void
cutout_kernel(const v4f* __restrict__ x,
              const int* __restrict__ cy,
              const int* __restrict__ cx,
              v4f* __restrict__ out)
{
    const int b  = blockIdx.y;                              // image (uniform)
    const int w4 = threadIdx.x;                             // 0..55
    const int r  = blockIdx.x * ROWS_PER_BLK + threadIdx.y; // 0..671 = c*224+h

    // h = r % 224, r in [0,672): at most two subtracts.
    int h = r;
    h = (h >= 224) ? h - 224 : h;
    h = (h >= 224) ? h - 224 : h;

    // Block-uniform hole center -> scalar loads (N_HOLES==1, layout [1,B]).
    const int cyb = cy[b];
    const int cxb = cx[b];

    // Uniform (SGPR) per-image bases + 32-bit per-lane element offset -> GVS.
    const v4f* __restrict__ xi = x   + (long)b * F4_PER_IMG;
    v4f*       __restrict__ oi = out + (long)b * F4_PER_IMG;
    const unsigned off = (unsigned)(r * W4_PER_ROW + w4);   // < 37632

    const bool in_y = (h >= cyb - LENGTH_HALF) & (h < cyb + LENGTH_HALF);
    const int  t0   = w4 * 4 - (cxb - LENGTH_HALF);         // w - lo, comp 0

    v4f v = xi[off];                                  // global_load_b128 (RT)

    // This float4 overlaps the box iff t0 in [-3, 15].
    if (in_y & ((unsigned)(t0 + 3) < 19u)) {
#pragma unroll
        for (int j = 0; j < 4; ++j) {
            const bool cov = (unsigned)(t0 + j) < 16u;
            v[j] = cov ? 0.0f : v[j];                 // v_cndmask_b32
        }
    }
    __builtin_nontemporal_store(v, &oi[off]);         // global_store_b128 th:TH_NT
}

extern "C" void kernel_launch(void* const* d_in, const int* in_sizes, int n_in,
                              void* d_out, int out_size, void* d_ws, size_t ws_size,
                              hipStream_t stream)
{
    (void)in_sizes; (void)n_in; (void)d_ws; (void)ws_size; (void)out_size;

    const v4f* x  = (const v4f*)d_in[0];   // [256,3,224,224] fp32 as float4
    const int* cy = (const int*)d_in[1];   // [1,256] int32
    const int* cx = (const int*)d_in[2];   // [1,256] int32
    v4f*       o  = (v4f*)d_out;

    const int B = 256;
    dim3 grid(ROWS_PER_IMG / ROWS_PER_BLK, B, 1);  // (168, 256)
    dim3 block(W4_PER_ROW, ROWS_PER_BLK, 1);       // 56x4 = 224 = 7 waves
    cutout_kernel<<<grid, block, 0, stream>>>(x, cy, cx, o);
}